// TripletLoss_42193758716072
// MI455X (gfx1250) — compile-verified
//
#include <hip/hip_runtime.h>
#include <math.h>

typedef __attribute__((ext_vector_type(2))) float v2f;
typedef __attribute__((ext_vector_type(8))) float v8f;

#define MAX_POS 96   // class size ~16 expected; generous safety margin
#define MAX_N   512

// ---------------------------------------------------------------------------
// Kernel 1: row squared norms. sqn[i] = sum_k feat[i][k]^2
// ---------------------------------------------------------------------------
__global__ void sqnorm_kernel(const float* __restrict__ feat,
                              float* __restrict__ sqn, int N, int d) {
  int i = blockIdx.x * blockDim.x + threadIdx.x;
  if (i >= N) return;
  const float4* row = (const float4*)(feat + (size_t)i * d);
  float s = 0.f;
  int n4 = d >> 2;
  for (int k = 0; k < n4; ++k) {
    float4 v = row[k];
    s = fmaf(v.x, v.x, s);
    s = fmaf(v.y, v.y, s);
    s = fmaf(v.z, v.z, s);
    s = fmaf(v.w, v.w, s);
  }
  sqn[i] = s;
}

// ---------------------------------------------------------------------------
// Kernel 2: D = sqrt(max(sqn_i + sqn_j - 2*G, eps)) with G = feat @ feat^T
// via V_WMMA_F32_16X16X4_F32. One wave32 per 16x16 output tile, K-loop in
// steps of 4. A-layout (16x4 f32): lanes 0-15 hold K=k0,k0+1 of row M=lane;
// lanes 16-31 hold K=k0+2,k0+3. B (4x16) mirrors with N=lane — since
// B = feat^T, the B load pattern equals the A pattern with the column base.
// ---------------------------------------------------------------------------
__global__ void gram_dist_wmma_kernel(const float* __restrict__ feat,
                                      const float* __restrict__ sqn,
                                      float* __restrict__ D, int N, int d) {
  const int lane = threadIdx.x & 31;
  const int wave = threadIdx.x >> 5;
  const int tilesPerRow = N >> 4;
  const int nTiles = tilesPerRow * tilesPerRow;
  const int tile = blockIdx.x * (blockDim.x >> 5) + wave;
  if (tile >= nTiles) return;               // wave-uniform: EXEC stays all-1s
  const int ti = tile / tilesPerRow;
  const int tj = tile % tilesPerRow;
  const int iBase = ti << 4, jBase = tj << 4;
  const int r = lane & 15;
  const int kOff = (lane >> 4) << 1;        // upper half-wave: K+2, K+3
  const float* __restrict__ aPtr = feat + (size_t)(iBase + r) * d + kOff;
  const float* __restrict__ bPtr = feat + (size_t)(jBase + r) * d + kOff;

  v8f c = {};
  for (int k0 = 0; k0 < d; k0 += 4) {
    v2f a, b;
    a.x = aPtr[k0];     a.y = aPtr[k0 + 1];
    b.x = bPtr[k0];     b.y = bPtr[k0 + 1];
    // (neg_a, A, neg_b, B, c_mod, C, reuse_a, reuse_b)
    c = __builtin_amdgcn_wmma_f32_16x16x4_f32(false, a, false, b,
                                              (short)0, c, false, false);
  }

  const int col = jBase + r;
  const float sc = sqn[col];
  const int rowOff = (lane >> 4) << 3;      // C/D: upper lanes hold M+8
#pragma unroll
  for (int v = 0; v < 8; ++v) {
    const int row = iBase + rowOff + v;
    const float sq = sqn[row] + sc - 2.0f * c[v];
    D[(size_t)row * N + col] = sqrtf(fmaxf(sq, 1e-12f));
  }
}

// ---------------------------------------------------------------------------
// Kernel 3: one block per anchor a. Compact the positive distances (serial
// on thread 0 for bit-determinism), then sweep negatives with 256 threads:
//   partSum[a] = sum_{p in P(a)} sum_{n in Neg(a)} sigmoid(10*(D[a,p]-D[a,n]))
//   partCnt[a] = |P(a)| * |Neg(a)|    (exact integer count)
// ---------------------------------------------------------------------------
__global__ void triplet_kernel(const int* __restrict__ y,
                               const float* __restrict__ D,
                               float* __restrict__ partSum,
                               int* __restrict__ partCnt, int N) {
  __shared__ int   yl[MAX_N];
  __shared__ float posD[MAX_POS];
  __shared__ float red[256];
  __shared__ int   shCnt[2];                // [0]=nPos, [1]=nNeg

  const int a   = blockIdx.x;
  const int tid = threadIdx.x;

  for (int i = tid; i < N; i += blockDim.x) yl[i] = y[i];
  __syncthreads();

  const int ya = yl[a];
  const float* __restrict__ Da = D + (size_t)a * N;

  if (tid == 0) {
    int np = 0, nn = 0;
    for (int p = 0; p < N; ++p) {
      if (yl[p] == ya) {
        if (p != a && np < MAX_POS) posD[np++] = Da[p];
      } else {
        ++nn;
      }
    }
    shCnt[0] = np;
    shCnt[1] = nn;
  }
  __syncthreads();

  const int nPos = shCnt[0];
  float acc = 0.f;
  for (int n = tid; n < N; n += blockDim.x) {
    if (yl[n] != ya) {
      const float dn = Da[n];
      for (int j = 0; j < nPos; ++j) {
        const float x = 10.0f * (posD[j] - dn);     // SCALING = 10
        acc += 1.0f / (1.0f + __expf(-x));          // sigmoid
      }
    }
  }
  red[tid] = acc;
  __syncthreads();
  for (int s = 128; s > 0; s >>= 1) {
    if (tid < s) red[tid] += red[tid + s];
    __syncthreads();
  }
  if (tid == 0) {
    partSum[a] = red[0];
    partCnt[a] = nPos * shCnt[1];
  }
}

// ---------------------------------------------------------------------------
// Kernel 4: deterministic single-block reduction -> out[0] = total / count
// ---------------------------------------------------------------------------
__global__ void finalize_kernel(const float* __restrict__ partSum,
                                const int* __restrict__ partCnt,
                                float* __restrict__ out, int N) {
  __shared__ float s[256];
  __shared__ int   c[256];
  const int t = threadIdx.x;
  float fs = 0.f;
  int   ic = 0;
  for (int i = t; i < N; i += 256) { fs += partSum[i]; ic += partCnt[i]; }
  s[t] = fs; c[t] = ic;
  __syncthreads();
  for (int k = 128; k > 0; k >>= 1) {
    if (t < k) { s[t] += s[t + k]; c[t] += c[t + k]; }
    __syncthreads();
  }
  if (t == 0) out[0] = s[0] / (float)c[0];
}

// ---------------------------------------------------------------------------
extern "C" void kernel_launch(void* const* d_in, const int* in_sizes, int n_in,
                              void* d_out, int out_size, void* d_ws, size_t ws_size,
                              hipStream_t stream) {
  const float* feat = (const float*)d_in[0];   // [N, d] f32
  // d_in[1] (logits) unused by the loss
  const int*   y    = (const int*)d_in[2];     // [N] i32
  float*       out  = (float*)d_out;           // scalar f32

  const int N = in_sizes[2];
  const int d = in_sizes[0] / N;

  // workspace layout
  char*  ws      = (char*)d_ws;
  float* sqn     = (float*)ws;                                   // N floats
  float* D       = (float*)(ws + 2048);                          // N*N floats
  size_t dBytes  = (size_t)N * N * sizeof(float);
  float* partSum = (float*)(ws + 2048 + dBytes);                 // N floats
  int*   partCnt = (int*)  (ws + 2048 + dBytes + 2048);          // N ints

  // 1) squared norms
  sqnorm_kernel<<<(N + 255) / 256, 256, 0, stream>>>(feat, sqn, N, d);

  // 2) WMMA distance matrix: 8 waves (tiles) per 256-thread block
  const int tilesPerRow = N / 16;
  const int nTiles = tilesPerRow * tilesPerRow;
  gram_dist_wmma_kernel<<<(nTiles + 7) / 8, 256, 0, stream>>>(feat, sqn, D, N, d);

  // 3) per-anchor triplet sums
  triplet_kernel<<<N, 256, 0, stream>>>(y, D, partSum, partCnt, N);

  // 4) final deterministic reduction
  finalize_kernel<<<1, 256, 0, stream>>>(partSum, partCnt, out, N);
}